// SparseAttentionFixed_36575941493070
// MI455X (gfx1250) — compile-verified
//
#include <hip/hip_runtime.h>
#include <hip/hip_bf16.h>
#include <stdint.h>

// SparseAttentionFixed mask generation for MI455X (gfx1250).
// Pure store-bandwidth problem (512 MiB f32 out, 0 FLOPs): stage the highly
// redundant row patterns in LDS once, then stream them to HBM with CDNA5
// async LDS->global b128 stores (ASYNCcnt path), marked non-temporal since
// the 512 MiB write-once stream is 2.7x the 192 MB L2.

#define S_DIM 2048
#define LBLK 45              // floor(sqrt(2048))
#define NEGV (-10000.0f)
#define ROWS_PER_BLOCK 16    // 16 consecutive rows span <= 2 local blocks (16 < 45)
#define THREADS 256          // 8 waves of wave32

// Non-temporal store hint: write-once stream, don't claim L2 residency.
#define ASYNC_ST_MODS " th:TH_STORE_NT"

// global_store_async_from_lds_b128: VADDR = 32-bit per-lane global byte offset,
// VSRC = LDS byte address, SADDR = uniform 64-bit base (GVS addressing).
__device__ __forceinline__ void async_store_b128(const float* __restrict__ base,
                                                 unsigned goff, unsigned loff) {
  asm volatile("global_store_async_from_lds_b128 %0, %1, %2" ASYNC_ST_MODS
               :
               : "v"(goff), "v"(loff), "s"(base)
               : "memory");
}

__global__ __launch_bounds__(THREADS)
void SparseAttentionFixed_mask_kernel(float* __restrict__ out) {
  __shared__ float patt[2][S_DIM];   // two staged 8 KB row patterns

  const int tid   = (int)threadIdx.x;
  const int row0g = (int)blockIdx.x * ROWS_PER_BLOCK;  // global row id in [0, 65536)
  const int slice = row0g / S_DIM;                     // (b*16 + h), 0..31
  const int r0    = row0g - slice * S_DIM;             // row within [S,S] slice
  const int h     = slice & 15;
  const bool is_local = (h < 8);

  const int blkA = r0 / LBLK;                          // local block of first row
  const int blkB = (r0 + ROWS_PER_BLOCK - 1) / LBLK;   // local block of last row

  if (is_local) {
    for (int c = tid; c < S_DIM; c += THREADS) {
      const int cb = c / LBLK;
      patt[0][c] = (cb == blkA) ? 0.0f : NEGV;
      patt[1][c] = (cb == blkB) ? 0.0f : NEGV;
    }
  } else {
    // global mask: one row pattern, identical for every row of this slice
    for (int c = tid; c < S_DIM; c += THREADS) {
      patt[0][c] = ((c % LBLK) == (LBLK - 1)) ? 0.0f : NEGV;
    }
  }
  __syncthreads();   // LDS patterns final before any async store reads them

  // LDS byte address of patt[0][0]: aperture rule -> low 32 bits of generic addr
  const unsigned lds0 = (unsigned)(uintptr_t)(&patt[0][0]);
  const unsigned lds1 = lds0 + (unsigned)(S_DIM * sizeof(float));
  const unsigned o0   = (unsigned)tid * 16u;             // this lane's chunk 0
  const unsigned o1   = o0 + (unsigned)THREADS * 16u;    // this lane's chunk 1

  // Byte offset of first row in the output tensor (max 512 MiB, fits u32).
  unsigned row_goff =
      (unsigned)(((size_t)slice * S_DIM + (size_t)r0) * (S_DIM * sizeof(float)));

  for (int i = 0; i < ROWS_PER_BLOCK; ++i) {
    const int r = r0 + i;
    const unsigned lrow =
        (is_local && ((r / LBLK) != blkA)) ? lds1 : lds0;
    // 2048 floats = 8192 B = 512 x 16B chunks; 256 threads -> 2 chunks each.
    async_store_b128(out, row_goff + o0, lrow + o0);
    async_store_b128(out, row_goff + o1, lrow + o1);
    row_goff += (unsigned)(S_DIM * sizeof(float));
  }
  // S_ENDPGM performs an implicit wait-idle, draining ASYNCcnt.
}

extern "C" void kernel_launch(void* const* d_in, const int* in_sizes, int n_in,
                              void* d_out, int out_size, void* d_ws, size_t ws_size,
                              hipStream_t stream) {
  (void)d_in; (void)in_sizes; (void)n_in; (void)d_ws; (void)ws_size; (void)out_size;
  float* out = (float*)d_out;

  const int total_rows = 2 * 16 * S_DIM;          // B*H*S = 65536 output rows
  dim3 grid(total_rows / ROWS_PER_BLOCK);         // 4096 workgroups
  dim3 block(THREADS);                            // 256 threads (8 x wave32)
  hipLaunchKernelGGL(SparseAttentionFixed_mask_kernel, grid, block, 0, stream, out);
}